// LTCCell_51153060496053
// MI455X (gfx1250) — compile-verified
//
#include <hip/hip_runtime.h>
#include <hip/hip_bf16.h>
#include <cstdint>

#define N_B      4096
#define N_IN     64
#define N_S      128
#define N_MOTOR  32
#define N_UNFOLD 6
#define LTC_EPS  1e-8f

// packed-parameter workspace layout (float indices)
#define PK_RE2S 0        // [128x128] -log2e * sigma
#define PK_RE2B 16384    // [128x128]  log2e * sigma * mu
#define PK_RWD  32768    // [128x128]  w * mask
#define PK_RWN  49152    // [128x128]  w * mask * erev
#define PK_SE2S 65536    // [64x128]  -log2e * s_sigma * input_w[i]
#define PK_SE2B 73728    // [64x128]   log2e * s_sigma * (s_mu - input_b[i])
#define PK_SWD  81920    // [64x128]   s_w * s_mask
#define PK_SWN  90112    // [64x128]   s_w * s_mask * s_erev
#define PK_CMT  98304    // [128] cm * 6
#define PK_GLVL 98432    // [128] gleak * vleak
#define PK_DENC 98560    // [128] cm*6 + gleak

#define AS1 __attribute__((address_space(1)))
#define AS3 __attribute__((address_space(3)))

typedef int v4i __attribute__((vector_size(16)));

// ---------------- CDNA5 async global->LDS staging ----------------
__device__ __forceinline__ void stage_b128(const float* __restrict__ g, float* l) {
#if __has_builtin(__builtin_amdgcn_global_load_async_to_lds_b128)
  __builtin_amdgcn_global_load_async_to_lds_b128((AS1 v4i*)g, (AS3 v4i*)l, 0, 0);
#else
  unsigned lds_off = (unsigned)(uintptr_t)(AS3 void*)l;
  asm volatile("global_load_async_to_lds_b128 %0, %1, off"
               :: "v"(lds_off), "v"(g) : "memory");
#endif
}

__device__ __forceinline__ void wait_async_lds() {
#if __has_builtin(__builtin_amdgcn_s_wait_asynccnt)
  __builtin_amdgcn_s_wait_asynccnt(0);
#else
  asm volatile("s_wait_asynccnt 0" ::: "memory");
#endif
}

// uniform-lane broadcast via v_readlane (SGPR result -> scalar FMA operand)
__device__ __forceinline__ float bcast(float x, int l) {
  return __int_as_float(__builtin_amdgcn_readlane(__float_as_int(x), l));
}

// 4 synapses (one presynaptic j, 4 postsynaptic s owned by this lane)
__device__ __forceinline__ void synapse4(const float* es, const float* eb,
                                         const float* wd, const float* wn,
                                         float vj, float4& den, float4& num) {
  const float4 E  = *(const float4*)es;
  const float4 Bv = *(const float4*)eb;
  const float4 D  = *(const float4*)wd;
  const float4 Nn = *(const float4*)wn;
  float g;
  g = __builtin_amdgcn_rcpf(1.0f + __builtin_amdgcn_exp2f(fmaf(E.x, vj, Bv.x)));
  den.x = fmaf(D.x, g, den.x); num.x = fmaf(Nn.x, g, num.x);
  g = __builtin_amdgcn_rcpf(1.0f + __builtin_amdgcn_exp2f(fmaf(E.y, vj, Bv.y)));
  den.y = fmaf(D.y, g, den.y); num.y = fmaf(Nn.y, g, num.y);
  g = __builtin_amdgcn_rcpf(1.0f + __builtin_amdgcn_exp2f(fmaf(E.z, vj, Bv.z)));
  den.z = fmaf(D.z, g, den.z); num.z = fmaf(Nn.z, g, num.z);
  g = __builtin_amdgcn_rcpf(1.0f + __builtin_amdgcn_exp2f(fmaf(E.w, vj, Bv.w)));
  den.w = fmaf(D.w, g, den.w); num.w = fmaf(Nn.w, g, num.w);
}

// ---------------- main kernel: 16 waves/block, 1 row per wave ----------------
// (defined first so the disasm snippet shows the async-staging prologue)
__global__ __launch_bounds__(512, 1) void ltc_main(
    const float* __restrict__ inputs, const float* __restrict__ states,
    const float* __restrict__ pk,     const float* __restrict__ output_w,
    const float* __restrict__ output_b, float* __restrict__ out) {
  extern __shared__ float smem[];  // 256 KB dynamic LDS (CDNA5 320KB/WGP)
  const int tid  = threadIdx.x;
  const int lane = tid & 31;
  const int wave = tid >> 5;
  const int sl   = lane * 4;                 // this lane owns s = sl..sl+3
  const int row  = blockIdx.x * 16 + wave;   // 256 blocks * 16 waves = 4096 rows

  // ---- phase 1: async-stage sensory params (32768 floats) into LDS ----
  // LDS: sE2S@0  sE2B@8192  sWD@16384  sWN@24576
#pragma unroll
  for (int c = 0; c < 16; ++c) {
    const int idx = (c * 512 + tid) * 4;
    stage_b128(pk + PK_SE2S + idx, smem + idx);
  }
  wait_async_lds();
  __syncthreads();

  const float xa = inputs[row * N_IN + lane];        // x[lane]
  const float xb = inputs[row * N_IN + 32 + lane];   // x[32+lane]
  float4 v = *(const float4*)(states + row * N_S + sl);

  float4 sden = {0.f, 0.f, 0.f, 0.f};
  float4 snum = {0.f, 0.f, 0.f, 0.f};
  for (int j = 0; j < 32; ++j) {
    const float vj = bcast(xa, j);
    synapse4(smem + j * 128 + sl,         smem + 8192  + j * 128 + sl,
             smem + 16384 + j * 128 + sl, smem + 24576 + j * 128 + sl,
             vj, sden, snum);
  }
  for (int j = 0; j < 32; ++j) {
    const float vj = bcast(xb, j);
    const int jj = j + 32;
    synapse4(smem + jj * 128 + sl,         smem + 8192  + jj * 128 + sl,
             smem + 16384 + jj * 128 + sl, smem + 24576 + jj * 128 + sl,
             vj, sden, snum);
  }
  __syncthreads();

  // ---- phase 2: async-stage recurrent params (65536 floats) into LDS ----
  // LDS: rE2S@0  rE2B@16384  rWD@32768  rWN@49152
#pragma unroll
  for (int c = 0; c < 32; ++c) {
    const int idx = (c * 512 + tid) * 4;
    stage_b128(pk + idx, smem + idx);
  }
  wait_async_lds();
  __syncthreads();

  const float4 cmt  = *(const float4*)(pk + PK_CMT  + sl);
  const float4 glvl = *(const float4*)(pk + PK_GLVL + sl);
  const float4 denc = *(const float4*)(pk + PK_DENC + sl);

  for (int it = 0; it < N_UNFOLD; ++it) {
    float4 den = sden, num = snum;
    for (int jj = 0; jj < N_S; jj += 4) {
      const int src = jj >> 2;               // lane owning v[jj..jj+3]
      const float va = bcast(v.x, src);
      const float vb = bcast(v.y, src);
      const float vc = bcast(v.z, src);
      const float vd = bcast(v.w, src);
      synapse4(smem + (jj+0)*128 + sl, smem + 16384 + (jj+0)*128 + sl,
               smem + 32768 + (jj+0)*128 + sl, smem + 49152 + (jj+0)*128 + sl, va, den, num);
      synapse4(smem + (jj+1)*128 + sl, smem + 16384 + (jj+1)*128 + sl,
               smem + 32768 + (jj+1)*128 + sl, smem + 49152 + (jj+1)*128 + sl, vb, den, num);
      synapse4(smem + (jj+2)*128 + sl, smem + 16384 + (jj+2)*128 + sl,
               smem + 32768 + (jj+2)*128 + sl, smem + 49152 + (jj+2)*128 + sl, vc, den, num);
      synapse4(smem + (jj+3)*128 + sl, smem + 16384 + (jj+3)*128 + sl,
               smem + 32768 + (jj+3)*128 + sl, smem + 49152 + (jj+3)*128 + sl, vd, den, num);
    }
    v.x = (fmaf(cmt.x, v.x, glvl.x) + num.x) / (denc.x + den.x + LTC_EPS);
    v.y = (fmaf(cmt.y, v.y, glvl.y) + num.y) / (denc.y + den.y + LTC_EPS);
    v.z = (fmaf(cmt.z, v.z, glvl.z) + num.z) / (denc.z + den.z + LTC_EPS);
    v.w = (fmaf(cmt.w, v.w, glvl.w) + num.w) / (denc.w + den.w + LTC_EPS);
  }

  // d_out = [output (4096x32) | v_pre (4096x128)]
  *(float4*)(out + N_B * N_MOTOR + row * N_S + sl) = v;
  if (lane < 8) {
    const float4 ow = *(const float4*)(output_w + sl);
    const float4 ob = *(const float4*)(output_b + sl);
    float4 o;
    o.x = fmaf(v.x, ow.x, ob.x);
    o.y = fmaf(v.y, ow.y, ob.y);
    o.z = fmaf(v.z, ow.z, ob.z);
    o.w = fmaf(v.w, ow.w, ob.w);
    *(float4*)(out + row * N_MOTOR + sl) = o;
  }
}

// ---------------- pack kernel: fold constants once ----------------
__global__ void ltc_pack(const float* __restrict__ gleak, const float* __restrict__ vleak,
                         const float* __restrict__ cm,    const float* __restrict__ w,
                         const float* __restrict__ sigma, const float* __restrict__ mu,
                         const float* __restrict__ erev,  const float* __restrict__ sw,
                         const float* __restrict__ ssigma,const float* __restrict__ smu,
                         const float* __restrict__ serev, const float* __restrict__ mask,
                         const float* __restrict__ smask, const float* __restrict__ iw,
                         const float* __restrict__ ib,    float* __restrict__ pk) {
  const float L2E = 1.4426950408889634f;
  const int t = blockIdx.x * blockDim.x + threadIdx.x;
  if (t < 16384) {  // recurrent [j=t>>7][s=t&127]
    const float sg = sigma[t], m = mu[t];
    const float we = w[t] * mask[t];
    pk[PK_RE2S + t] = -L2E * sg;
    pk[PK_RE2B + t] =  L2E * sg * m;
    pk[PK_RWD  + t] = we;
    pk[PK_RWN  + t] = we * erev[t];
  }
  if (t < 8192) {   // sensory [i=t>>7][s=t&127]; fold input affine x=in*iw+ib
    const int i = t >> 7;
    const float sg = ssigma[t], m = smu[t];
    const float we = sw[t] * smask[t];
    pk[PK_SE2S + t] = -L2E * sg * iw[i];
    pk[PK_SE2B + t] =  L2E * sg * (m - ib[i]);
    pk[PK_SWD  + t] = we;
    pk[PK_SWN  + t] = we * serev[t];
  }
  if (t < 128) {
    const float c = cm[t] * (float)N_UNFOLD;  // cm / (1.0 / 6)
    pk[PK_CMT  + t] = c;
    pk[PK_GLVL + t] = gleak[t] * vleak[t];
    pk[PK_DENC + t] = c + gleak[t];
  }
}

extern "C" void kernel_launch(void* const* d_in, const int* in_sizes, int n_in,
                              void* d_out, int out_size, void* d_ws, size_t ws_size,
                              hipStream_t stream) {
  (void)in_sizes; (void)n_in; (void)out_size; (void)ws_size;
  const float* inputs = (const float*)d_in[0];
  const float* states = (const float*)d_in[1];
  const float* gleak  = (const float*)d_in[2];
  const float* vleak  = (const float*)d_in[3];
  const float* cm     = (const float*)d_in[4];
  const float* w      = (const float*)d_in[5];
  const float* sigma  = (const float*)d_in[6];
  const float* mu     = (const float*)d_in[7];
  const float* erev   = (const float*)d_in[8];
  const float* sw     = (const float*)d_in[9];
  const float* ssig   = (const float*)d_in[10];
  const float* smu    = (const float*)d_in[11];
  const float* serev  = (const float*)d_in[12];
  const float* mask   = (const float*)d_in[13];
  const float* smask  = (const float*)d_in[14];
  const float* iw     = (const float*)d_in[15];
  const float* ib     = (const float*)d_in[16];
  const float* ow     = (const float*)d_in[17];
  const float* ob     = (const float*)d_in[18];
  float* pk  = (float*)d_ws;
  float* out = (float*)d_out;

  (void)hipFuncSetAttribute(reinterpret_cast<const void*>(ltc_main),
                            hipFuncAttributeMaxDynamicSharedMemorySize, 262144);

  ltc_pack<<<64, 256, 0, stream>>>(gleak, vleak, cm, w, sigma, mu, erev,
                                   sw, ssig, smu, serev, mask, smask, iw, ib, pk);
  ltc_main<<<256, 512, 262144, stream>>>(inputs, states, pk, ow, ob, out);
}